// GptOssTopKRouterNew_29394756173987
// MI455X (gfx1250) — compile-verified
//
#include <hip/hip_runtime.h>

typedef __attribute__((ext_vector_type(2))) float v2f;
typedef __attribute__((ext_vector_type(8))) float v8f;

#define HDIM 2880
#define NEXP 32
#define NTOK 16384
#define TOPK 4

// ---------------------------------------------------------------------------
// Kernel 1: router GEMM  logits[N,32] = x[N,2880] @ W[32,2880]^T + bias
// One wave32 computes a 16-token x 32-expert tile using V_WMMA_F32_16X16X4_F32.
//
// A layout (16x4 f32, ISA 7.12.2): lane L holds row M = L%16; VGPR v holds
// K = 2*(L/16) + v.  A float2 load at (row, k0 + 2*(L/16)) lands exactly in
// the required {VGPR0, VGPR1} positions for both lane halves.
// B layout mirrors A with N = L%16 (expert), K = 2*(L/16) + v, so a float2
// load from the weight row works identically (weight is [expert][H] row-major).
// C/D layout: lane L, VGPR v -> element (M = 8*(L/16) + v, N = L%16).
// ---------------------------------------------------------------------------
__global__ __launch_bounds__(128)
void router_gemm_kernel(const float* __restrict__ x,
                        const float* __restrict__ w,
                        const float* __restrict__ bias,
                        float* __restrict__ logits)
{
    const int lane      = threadIdx.x & 31;
    const int waveInBlk = threadIdx.x >> 5;
    const int tile      = blockIdx.x * 4 + waveInBlk;   // 16-token tile id
    const int tokBase   = tile * 16;

    const int rowSel = lane & 15;         // A: token-in-tile ; B: expert-in-half
    const int kSel   = (lane >> 4) * 2;   // 0 for lanes 0-15, 2 for lanes 16-31

    const float* aPtr  = x + (size_t)(tokBase + rowSel) * HDIM + kSel;
    const float* bPtr0 = w + (size_t)rowSel        * HDIM + kSel;   // experts 0..15
    const float* bPtr1 = w + (size_t)(rowSel + 16) * HDIM + kSel;   // experts 16..31

    v8f acc0 = {};
    v8f acc1 = {};

    for (int k0 = 0; k0 < HDIM; k0 += 16) {
#pragma unroll
        for (int s = 0; s < 4; ++s) {
            const int k = k0 + s * 4;
            v2f a  = *(const v2f*)(aPtr  + k);
            v2f b0 = *(const v2f*)(bPtr0 + k);
            v2f b1 = *(const v2f*)(bPtr1 + k);
            // (neg_a, A, neg_b, B, c_mod, C, reuse_a, reuse_b)
            acc0 = __builtin_amdgcn_wmma_f32_16x16x4_f32(
                       false, a, false, b0, (short)0, acc0, false, false);
            acc1 = __builtin_amdgcn_wmma_f32_16x16x4_f32(
                       false, a, false, b1, (short)0, acc1, false, false);
        }
    }

    const float bias0 = bias[rowSel];
    const float bias1 = bias[rowSel + 16];
    const int   mBase = (lane >> 4) * 8;

#pragma unroll
    for (int v = 0; v < 8; ++v) {
        const int tok = tokBase + mBase + v;
        logits[(size_t)tok * NEXP + rowSel]      = acc0[v] + bias0;
        logits[(size_t)tok * NEXP + rowSel + 16] = acc1[v] + bias1;
    }
}

// ---------------------------------------------------------------------------
// Kernel 2: per-token top-4 -> softmax over selected -> dense scatter.
// One thread per token; logits are read from the scores region of d_out and
// overwritten in place. Fully register-resident: argmax passes use a selected-
// bitmask (compile-time bit tests) and the scatter uses select chains, so no
// dynamic VGPR indexing / scratch.
// ---------------------------------------------------------------------------
__global__ __launch_bounds__(256)
void topk_softmax_kernel(float* __restrict__ scores, int* __restrict__ idx_out)
{
    const int tok = blockIdx.x * blockDim.x + threadIdx.x;
    if (tok >= NTOK) return;

    float l[NEXP];
#pragma unroll
    for (int e = 0; e < NEXP; ++e) l[e] = scores[(size_t)tok * NEXP + e];

    int      idx[TOPK];
    float    val[TOPK];
    unsigned sel = 0u;

#pragma unroll
    for (int k = 0; k < TOPK; ++k) {
        int   best = -1;
        float bv   = -__builtin_huge_valf();
#pragma unroll
        for (int e = 0; e < NEXP; ++e) {
            const bool avail = ((sel >> e) & 1u) == 0u;
            const bool take  = avail && (l[e] > bv);   // strict > : ties -> lowest index
            bv   = take ? l[e] : bv;
            best = take ? e    : best;
        }
        idx[k] = best;
        val[k] = bv;
        sel |= (1u << best);
    }

    // softmax over the 4 selected logits (val[0] is the max)
    const float m = val[0];
    float wgt[TOPK];
    float s = 0.f;
#pragma unroll
    for (int k = 0; k < TOPK; ++k) { wgt[k] = __expf(val[k] - m); s += wgt[k]; }
    const float inv = 1.f / s;

#pragma unroll
    for (int e = 0; e < NEXP; ++e) {
        float o = 0.f;
#pragma unroll
        for (int k = 0; k < TOPK; ++k) o = (idx[k] == e) ? wgt[k] * inv : o;
        scores[(size_t)tok * NEXP + e] = o;
    }
#pragma unroll
    for (int k = 0; k < TOPK; ++k)
        idx_out[(size_t)tok * TOPK + k] = idx[k];
}

// ---------------------------------------------------------------------------
extern "C" void kernel_launch(void* const* d_in, const int* in_sizes, int n_in,
                              void* d_out, int out_size, void* d_ws, size_t ws_size,
                              hipStream_t stream)
{
    const float* x    = (const float*)d_in[0];  // [4,4096,2880] f32
    const float* w    = (const float*)d_in[1];  // [32,2880]     f32
    const float* bias = (const float*)d_in[2];  // [32]          f32

    float* scores  = (float*)d_out;                         // [16384,32] f32
    int*   top_idx = (int*)(scores + (size_t)NTOK * NEXP);  // [16384,4]  i32

    // GEMM: 1024 tiles of 16 tokens, 4 waves (128 threads) per block.
    router_gemm_kernel<<<dim3(NTOK / 16 / 4), dim3(128), 0, stream>>>(x, w, bias, scores);

    // Top-k + softmax + scatter, in place on the scores block.
    topk_softmax_kernel<<<dim3(NTOK / 256), dim3(256), 0, stream>>>(scores, top_idx);
}